// SubGraph2_70600672412044
// MI455X (gfx1250) — compile-verified
//
#include <hip/hip_runtime.h>

typedef __attribute__((ext_vector_type(2))) float v2f;
typedef __attribute__((ext_vector_type(8))) float v8f;

#define HID   64
#define NSEG  100000
#define NROWS 2000000
#define EPS_LN 1e-5f

__device__ __forceinline__ float half_wave_sum(float v) {
    // butterfly over 16-lane groups (wave32; xor masks <= 8 stay in-group)
    v += __shfl_xor(v, 1, 32);
    v += __shfl_xor(v, 2, 32);
    v += __shfl_xor(v, 4, 32);
    v += __shfl_xor(v, 8, 32);
    return v;
}

__global__ void zero_kernel(float* __restrict__ p, int n) {
    int i = blockIdx.x * blockDim.x + threadIdx.x;
    int stride = gridDim.x * blockDim.x;
    for (; i < n; i += stride) p[i] = 0.0f;
}

// Shared epilogue: bias + LayerNorm(64) + ReLU, store h row-major, atomicMax into agg.
// acc[j] holds the 16x16 tile for columns [16j, 16j+16). C/D layout: VGPR r,
// lanes 0-15 -> row (base+r), lanes 16-31 -> row (base+8+r), N = lane&15.
__device__ __forceinline__ void mlp_epilogue(
    v8f acc[4], const float bcol[4], const float gcol[4], const float becol[4],
    int base, int lane, const int* __restrict__ cat,
    float* __restrict__ h_out, unsigned int* __restrict__ agg)
{
    const int hl   = lane & 15;
    const int rofs = (lane >> 4) << 3;   // 0 or 8
#pragma unroll
    for (int r = 0; r < 8; ++r) {
        float t[4];
        float s = 0.0f, q = 0.0f;
#pragma unroll
        for (int j = 0; j < 4; ++j) {
            float v = acc[j][r] + bcol[j];
            t[j] = v; s += v; q += v * v;
        }
        s = half_wave_sum(s);
        q = half_wave_sum(q);
        float mu  = s * (1.0f / 64.0f);
        float var = q * (1.0f / 64.0f) - mu * mu;
        float rs  = rsqrtf(var + EPS_LN);
        int row = base + r + rofs;
        int catr = cat[row];
        float*        hr = h_out + row  * HID;
        unsigned int* ar = agg   + catr * HID;
#pragma unroll
        for (int j = 0; j < 4; ++j) {
            int col = j * 16 + hl;
            float v = (t[j] - mu) * rs * gcol[j] + becol[j];
            v = fmaxf(v, 0.0f);                           // ReLU -> v >= 0
            hr[col] = v;
            // agg zero-initialized; 0 is the identity, skip ~half the atomics
            if (v > 0.0f)
                atomicMax(ar + col, __float_as_uint(v));  // bit-pattern max valid for v>=0
        }
    }
}

// Layer 0: din = 8. x is [N,8] row-major. B fragments fit in registers.
__global__ void layer0_kernel(
    const float* __restrict__ x, const int* __restrict__ cat,
    const float* __restrict__ W, const float* __restrict__ b,
    const float* __restrict__ g, const float* __restrict__ be,
    float* __restrict__ h_out, unsigned int* __restrict__ agg, int ntiles)
{
    const int tid    = threadIdx.x;
    const int lane   = tid & 31;
    const int hl     = lane & 15;
    const int koff   = (lane >> 4) << 1;  // A/B frag K sub-offset: 0 or 2
    const int wave   = (blockIdx.x * blockDim.x + tid) >> 5;
    const int nwaves = (gridDim.x * blockDim.x) >> 5;

    // B fragments (4x16 of W) in registers: k-chunks 0..1, col-tiles j 0..3
    v2f bfrag[2][4];
#pragma unroll
    for (int k = 0; k < 2; ++k)
#pragma unroll
        for (int j = 0; j < 4; ++j) {
            int col  = j * 16 + hl;
            int krow = 4 * k + koff;
            v2f w;
            w.x = W[krow * HID + col];
            w.y = W[(krow + 1) * HID + col];
            bfrag[k][j] = w;
        }
    float bcol[4], gcol[4], becol[4];
#pragma unroll
    for (int j = 0; j < 4; ++j) {
        int col = j * 16 + hl;
        bcol[j] = b[col]; gcol[j] = g[col]; becol[j] = be[col];
    }

    const v8f zero = {0.f,0.f,0.f,0.f,0.f,0.f,0.f,0.f};
    for (int tile = wave; tile < ntiles; tile += nwaves) {
        int base = tile * 16;
        const float* xr = x + (base + hl) * 8;  // A layout: lane&15 selects row M
        v8f acc[4] = {zero, zero, zero, zero};
#pragma unroll
        for (int k = 0; k < 2; ++k) {
            v2f a = *(const v2f*)(xr + 4 * k + koff);
#pragma unroll
            for (int j = 0; j < 4; ++j)
                acc[j] = __builtin_amdgcn_wmma_f32_16x16x4_f32(
                    false, a, false, bfrag[k][j], (short)0, acc[j], false, false);
        }
        mlp_epilogue(acc, bcol, gcol, becol, base, lane, cat, h_out, agg);
    }
}

// Layers 1,2: din = 128 = concat(h_prev[row], agg_prev[cat[row]]).
// W (128x64 = 32KB) staged to LDS in WMMA B-fragment layout -> ds_load_b64 in loop.
__global__ void layerN_kernel(
    const float* __restrict__ h_prev, const float* __restrict__ aggf_prev,
    const int* __restrict__ cat,
    const float* __restrict__ W, const float* __restrict__ b,
    const float* __restrict__ g, const float* __restrict__ be,
    float* __restrict__ h_out, unsigned int* __restrict__ agg, int ntiles)
{
    __shared__ v2f ldsFrag[32 * 4 * 32];  // kchunk * jtile * lane, 32KB

    const int tid = threadIdx.x;
    for (int idx = tid; idx < 32 * 4 * 32; idx += blockDim.x) {
        int l    = idx & 31;
        int j    = (idx >> 5) & 3;
        int k    = idx >> 7;
        int col  = j * 16 + (l & 15);
        int krow = 4 * k + ((l >> 4) << 1);
        v2f w;
        w.x = W[krow * HID + col];
        w.y = W[(krow + 1) * HID + col];
        ldsFrag[idx] = w;
    }
    __syncthreads();

    const int lane   = tid & 31;
    const int hl     = lane & 15;
    const int koff   = (lane >> 4) << 1;
    const int wave   = (blockIdx.x * blockDim.x + tid) >> 5;
    const int nwaves = (gridDim.x * blockDim.x) >> 5;

    float bcol[4], gcol[4], becol[4];
#pragma unroll
    for (int j = 0; j < 4; ++j) {
        int col = j * 16 + hl;
        bcol[j] = b[col]; gcol[j] = g[col]; becol[j] = be[col];
    }

    const v8f zero = {0.f,0.f,0.f,0.f,0.f,0.f,0.f,0.f};
    for (int tile = wave; tile < ntiles; tile += nwaves) {
        int base = tile * 16;
        int arow = base + hl;
        const float* hr = h_prev + arow * HID;
        const float* ar = aggf_prev + cat[arow] * HID;
        v8f acc[4] = {zero, zero, zero, zero};
        // K = 0..63 from h_prev
#pragma unroll
        for (int k = 0; k < 16; ++k) {
            v2f a = *(const v2f*)(hr + 4 * k + koff);
            const v2f* bf = &ldsFrag[(k * 4) * 32 + lane];
#pragma unroll
            for (int j = 0; j < 4; ++j)
                acc[j] = __builtin_amdgcn_wmma_f32_16x16x4_f32(
                    false, a, false, bf[j * 32], (short)0, acc[j], false, false);
        }
        // K = 64..127 from agg_prev[cat]
#pragma unroll
        for (int k = 16; k < 32; ++k) {
            v2f a = *(const v2f*)(ar + 4 * (k - 16) + koff);
            const v2f* bf = &ldsFrag[(k * 4) * 32 + lane];
#pragma unroll
            for (int j = 0; j < 4; ++j)
                acc[j] = __builtin_amdgcn_wmma_f32_16x16x4_f32(
                    false, a, false, bf[j * 32], (short)0, acc[j], false, false);
        }
        mlp_epilogue(acc, bcol, gcol, becol, base, lane, cat, h_out, agg);
    }
}

// Pass 1 of the tail: per-column sum of squares of x3 = concat(h2, agg2[cat]).
// Does NOT write x3 (recomputed in pass 2) -- saves a 1GB round-trip.
__global__ void sumsq_kernel(
    const float* __restrict__ h2, const float* __restrict__ agg2f,
    const int* __restrict__ cat,
    float* __restrict__ sumsq, int total4)   // total4 = N*128/4
{
    __shared__ float bins[128];
    int tid = threadIdx.x;
    for (int i = tid; i < 128; i += blockDim.x) bins[i] = 0.0f;
    __syncthreads();
    int stride = gridDim.x * blockDim.x;
    for (int e = blockIdx.x * blockDim.x + tid; e < total4; e += stride) {
        int row = e >> 5;             // 32 float4 per row of 128
        int c0  = (e & 31) << 2;      // column of first element (mult of 4)
        float4 v;
        if (c0 < 64)
            v = *(const float4*)(h2 + row * HID + c0);
        else
            v = *(const float4*)(agg2f + cat[row] * HID + (c0 - 64));
        atomicAdd(&bins[c0 + 0], v.x * v.x);
        atomicAdd(&bins[c0 + 1], v.y * v.y);
        atomicAdd(&bins[c0 + 2], v.z * v.z);
        atomicAdd(&bins[c0 + 3], v.w * v.w);
    }
    __syncthreads();
    for (int i = tid; i < 128; i += blockDim.x) atomicAdd(&sumsq[i], bins[i]);
}

// Pass 2: recompute x3 (h2 read + L2-resident agg gather), scale, single write to out.
__global__ void concat_scale_kernel(
    const float* __restrict__ h2, const float* __restrict__ agg2f,
    const int* __restrict__ cat, const float* __restrict__ sumsq,
    float* __restrict__ out, int total4)
{
    __shared__ float inv[128];
    int tid = threadIdx.x;
    for (int i = tid; i < 128; i += blockDim.x)
        inv[i] = 1.0f / (sqrtf(sumsq[i]) + 1e-12f);
    __syncthreads();
    int stride = gridDim.x * blockDim.x;
    for (int e = blockIdx.x * blockDim.x + tid; e < total4; e += stride) {
        int row = e >> 5;
        int c0  = (e & 31) << 2;
        float4 v;
        if (c0 < 64)
            v = *(const float4*)(h2 + row * HID + c0);
        else
            v = *(const float4*)(agg2f + cat[row] * HID + (c0 - 64));
        v.x *= inv[c0 + 0]; v.y *= inv[c0 + 1]; v.z *= inv[c0 + 2]; v.w *= inv[c0 + 3];
        *(float4*)(out + ((long long)e << 2)) = v;
    }
}

extern "C" void kernel_launch(void* const* d_in, const int* in_sizes, int n_in,
                              void* d_out, int out_size, void* d_ws, size_t ws_size,
                              hipStream_t stream) {
    const float* x   = (const float*)d_in[0];
    const int*   cat = (const int*)  d_in[1];
    // d_in[2] = num_segments (known: 100000)
    const float* W0  = (const float*)d_in[3];
    const float* b0  = (const float*)d_in[4];
    const float* g0  = (const float*)d_in[5];
    const float* be0 = (const float*)d_in[6];
    const float* W1  = (const float*)d_in[7];
    const float* b1  = (const float*)d_in[8];
    const float* g1  = (const float*)d_in[9];
    const float* be1 = (const float*)d_in[10];
    const float* W2  = (const float*)d_in[11];
    const float* b2  = (const float*)d_in[12];
    const float* g2  = (const float*)d_in[13];
    const float* be2 = (const float*)d_in[14];

    float* ws    = (float*)d_ws;
    float* hA    = ws;                                   // [N,64]
    float* hB    = hA   + (size_t)NROWS * HID;           // [N,64]
    float* agg0  = hB   + (size_t)NROWS * HID;           // [NSEG,64]
    float* agg1  = agg0 + (size_t)NSEG  * HID;           // [NSEG,64]
    float* sumsq = agg1 + (size_t)NSEG  * HID;           // [128]

    const int ntiles = NROWS / 16;
    const int total4 = NROWS * 128 / 4;

    zero_kernel<<<512, 256, 0, stream>>>(agg0, NSEG * HID);
    zero_kernel<<<512, 256, 0, stream>>>(agg1, NSEG * HID);
    zero_kernel<<<1, 128, 0, stream>>>(sumsq, 128);

    layer0_kernel<<<2048, 256, 0, stream>>>(x, cat, W0, b0, g0, be0,
                                            hA, (unsigned int*)agg0, ntiles);
    layerN_kernel<<<2048, 256, 0, stream>>>(hA, agg0, cat, W1, b1, g1, be1,
                                            hB, (unsigned int*)agg1, ntiles);
    zero_kernel<<<512, 256, 0, stream>>>(agg0, NSEG * HID);   // reuse as agg2
    layerN_kernel<<<2048, 256, 0, stream>>>(hB, agg1, cat, W2, b2, g2, be2,
                                            hA, (unsigned int*)agg0, ntiles); // hA = h2
    sumsq_kernel<<<2048, 256, 0, stream>>>(hA, agg0, cat, sumsq, total4);
    concat_scale_kernel<<<2048, 256, 0, stream>>>(hA, agg0, cat, sumsq,
                                                  (float*)d_out, total4);
}